// PointNetSetAbstraction_11123965297216
// MI455X (gfx1250) — compile-verified
//
#include <hip/hip_runtime.h>
#include <cstddef>

// ---------------- problem constants ----------------
#define Bb     16
#define Nn     8192
#define Ss     2048
#define Kk     32
#define MIDC   64
#define OUTC   128
#define EPSf   1e-5f
#define ROWS   (Bb*Ss*Kk)          // 1,048,576 rows of the MLP
#define NSTRIP (ROWS/16)           // 65,536 16-row WMMA strips

// workspace layout (float offsets)
#define XWS_FLOATS (4u*ROWS)       // grouped_xyz_norm, padded to 4 floats/row
#define iM     0                   // 9 moment accumulators
#define iSUM2  16                  // 128 ch sums of conv2 pre-act
#define iSSQ2  144                 // 128 ch sumsq
#define iW1F   288                 // folded conv1 weights (64x3)
#define iB1F   480                 // folded conv1 bias (64)
#define iA2    544                 // folded BN2 scale (128)
#define iB2F   672                 // folded BN2 shift (128)
#define FEAT_OFF (Bb*Ss*3)         // feat starts after new_xyz in d_out

typedef float v2f __attribute__((ext_vector_type(2)));
typedef float v8f __attribute__((ext_vector_type(8)));

// ---------------- K0: zero the stat region ----------------
__global__ void k_zero(float* stat) {
    if (threadIdx.x < 1024) stat[threadIdx.x] = 0.0f;
}

// ---------------- K1: farthest point sampling ----------------
// one block per batch; xyz[b] resident in LDS; wave32 shfl argmax reduction.
__global__ void k_fps(const float* __restrict__ xyz, float* __restrict__ out_newxyz) {
    extern __shared__ float sm[];
    float* sx = sm;
    float* sy = sm + Nn;
    float* sz = sm + 2*Nn;
    float* wval  = sm + 3*Nn;            // 32
    int*   widx  = (int*)(sm + 3*Nn + 32);
    int*   bcast = (int*)(sm + 3*Nn + 64);

    const int b = blockIdx.x, tid = threadIdx.x;
    const float* base = xyz + (size_t)b*Nn*3;
    for (int i = tid; i < Nn; i += 1024) {
        sx[i] = base[i*3+0]; sy[i] = base[i*3+1]; sz[i] = base[i*3+2];
    }
    __syncthreads();

    float dist[8];
    int   pts[8];
#pragma unroll
    for (int j = 0; j < 8; ++j) { pts[j] = tid + j*1024; dist[j] = 1e10f; }

    const int lane = tid & 31, wv = tid >> 5;
    int farthest = 0;

    for (int s = 0; s < Ss; ++s) {
        float cx = sx[farthest], cy = sy[farthest], cz = sz[farthest];
        if (tid == 0) {
            float* o = out_newxyz + ((size_t)b*Ss + s)*3;
            o[0] = cx; o[1] = cy; o[2] = cz;
        }
        float bv = -1.0f; int bi = Nn;
#pragma unroll
        for (int j = 0; j < 8; ++j) {
            float dx = sx[pts[j]] - cx, dy = sy[pts[j]] - cy, dz = sz[pts[j]] - cz;
            float d = dx*dx + dy*dy + dz*dz;
            dist[j] = fminf(dist[j], d);
            if (dist[j] > bv || (dist[j] == bv && pts[j] < bi)) { bv = dist[j]; bi = pts[j]; }
        }
        // wave32 reduce: max value, min index on ties (matches jnp.argmax)
#pragma unroll
        for (int m = 16; m > 0; m >>= 1) {
            float ov = __shfl_xor(bv, m, 32);
            int   oi = __shfl_xor(bi, m, 32);
            if (ov > bv || (ov == bv && oi < bi)) { bv = ov; bi = oi; }
        }
        if (lane == 0) { wval[wv] = bv; widx[wv] = bi; }
        __syncthreads();
        if (wv == 0) {
            bv = wval[lane]; bi = widx[lane];
#pragma unroll
            for (int m = 16; m > 0; m >>= 1) {
                float ov = __shfl_xor(bv, m, 32);
                int   oi = __shfl_xor(bi, m, 32);
                if (ov > bv || (ov == bv && oi < bi)) { bv = ov; bi = oi; }
            }
            if (lane == 0) bcast[0] = bi;
        }
        __syncthreads();
        farthest = bcast[0];
    }
}

// ---------------- K2: kNN (top-32 smallest dist) + grouping ----------------
// thread per query; full xyz[b] in LDS; register insertion sort (stable ties).
__global__ void k_knn_group(const float* __restrict__ xyz,
                            const float* __restrict__ newxyz,
                            float* __restrict__ xws) {
    extern __shared__ float sm[];
    float* sx = sm; float* sy = sm + Nn; float* sz = sm + 2*Nn;
    const int b = blockIdx.y;
    const int s = blockIdx.x*blockDim.x + threadIdx.x;
    const float* base = xyz + (size_t)b*Nn*3;
    for (int i = threadIdx.x; i < Nn; i += blockDim.x) {
        sx[i] = base[i*3+0]; sy[i] = base[i*3+1]; sz[i] = base[i*3+2];
    }
    __syncthreads();

    const float* q = newxyz + ((size_t)b*Ss + s)*3;
    const float qx = q[0], qy = q[1], qz = q[2];

    float bd[Kk]; int bi[Kk];
#pragma unroll
    for (int k = 0; k < Kk; ++k) { bd[k] = 3.0e38f; bi[k] = 0; }

    for (int i = 0; i < Nn; ++i) {
        float dx = sx[i]-qx, dy = sy[i]-qy, dz = sz[i]-qz;
        float d = dx*dx + dy*dy + dz*dz;
        if (d < bd[Kk-1]) {
            int j = Kk-1;
            while (j > 0 && bd[j-1] > d) { bd[j] = bd[j-1]; bi[j] = bi[j-1]; --j; }
            bd[j] = d; bi[j] = i;
        }
    }
    float* xo = xws + (((size_t)b*Ss + s)*Kk)*4;   // padded rows (4 floats)
#pragma unroll
    for (int k = 0; k < Kk; ++k) {
        int i = bi[k];
        float4 v = make_float4(sx[i]-qx, sy[i]-qy, sz[i]-qz, 0.0f);
        *(float4*)(xo + k*4) = v;
    }
}

// ---------------- K3: first/second moments of x (for exact BN1 stats) ----------------
__global__ void k_moments(const float* __restrict__ xws, float* __restrict__ stat) {
    float a0=0,a1=0,a2=0,a3=0,a4=0,a5=0,a6=0,a7=0,a8=0;
    for (size_t r = blockIdx.x*blockDim.x + threadIdx.x; r < (size_t)ROWS;
         r += (size_t)gridDim.x*blockDim.x) {
        float4 v = *(const float4*)(xws + r*4);
        a0 += v.x; a1 += v.y; a2 += v.z;
        a3 += v.x*v.x; a4 += v.x*v.y; a5 += v.x*v.z;
        a6 += v.y*v.y; a7 += v.y*v.z; a8 += v.z*v.z;
    }
    float acc[9] = {a0,a1,a2,a3,a4,a5,a6,a7,a8};
#pragma unroll
    for (int i = 0; i < 9; ++i) {
        float v = acc[i];
#pragma unroll
        for (int m = 16; m > 0; m >>= 1) v += __shfl_xor(v, m, 32);
        if ((threadIdx.x & 31) == 0) atomicAdd(&stat[iM + i], v);
    }
}

// ---------------- K3b: fold BN1 into conv1 (b1 cancels exactly) ----------------
__global__ void k_fin1(const float* __restrict__ W1, const float* __restrict__ g1,
                       const float* __restrict__ be1, float* __restrict__ stat) {
    int c = threadIdx.x;
    if (c >= MIDC) return;
    const float Rinv = 1.0f / (float)ROWS;
    float Ex = stat[iM+0]*Rinv, Ey = stat[iM+1]*Rinv, Ez = stat[iM+2]*Rinv;
    float Cxx = stat[iM+3]*Rinv - Ex*Ex, Cxy = stat[iM+4]*Rinv - Ex*Ey;
    float Cxz = stat[iM+5]*Rinv - Ex*Ez, Cyy = stat[iM+6]*Rinv - Ey*Ey;
    float Cyz = stat[iM+7]*Rinv - Ey*Ez, Czz = stat[iM+8]*Rinv - Ez*Ez;
    float wx = W1[c*3], wy = W1[c*3+1], wz = W1[c*3+2];
    float var = wx*wx*Cxx + wy*wy*Cyy + wz*wz*Czz
              + 2.0f*(wx*wy*Cxy + wx*wz*Cxz + wy*wz*Cyz);
    float A = g1[c] * rsqrtf(var + EPSf);
    stat[iW1F + c*3 + 0] = wx*A;
    stat[iW1F + c*3 + 1] = wy*A;
    stat[iW1F + c*3 + 2] = wz*A;
    stat[iB1F + c] = be1[c] - (wx*Ex + wy*Ey + wz*Ez)*A;
}

// -------- shared helper: build fp32 A-fragment (16x4 x 16 K-steps) on the fly --------
// Regenerates h1 = relu(BN1(conv1(x))) in fragment layout: lane l<16 -> K={4j,4j+1},
// lane l>=16 -> K={4j+2,4j+3}, row = r0 + (l&15).
__device__ __forceinline__ void make_afrag(const float* __restrict__ xws, int r0, int lane,
                                           const float* __restrict__ W1Fs,
                                           const float* __restrict__ B1Fs,
                                           v2f (&af)[16]) {
    const int row = r0 + (lane & 15);
    float4 xv = *(const float4*)(xws + (size_t)row*4);
    const int half = (lane >> 4) << 1;   // 0 or 2
#pragma unroll
    for (int j = 0; j < 16; ++j) {
        int c0 = j*4 + half;
        float h0 = fmaxf(xv.x*W1Fs[c0*3] + xv.y*W1Fs[c0*3+1] + xv.z*W1Fs[c0*3+2] + B1Fs[c0], 0.0f);
        int c1 = c0 + 1;
        float h1 = fmaxf(xv.x*W1Fs[c1*3] + xv.y*W1Fs[c1*3+1] + xv.z*W1Fs[c1*3+2] + B1Fs[c1], 0.0f);
        af[j] = v2f{h0, h1};
    }
}

// -------- register-resident B-fragments for one 16-channel N-tile --------
// bf[j] = {W2[n][k], W2[n][k+1]}, n = t*16+(lane&15), k = 4j + 2*(lane>=16).
// 16 x v2f = 32 VGPRs per tile; loaded once per wave (W2 is L2-resident).
__device__ __forceinline__ void load_bfrag(const float* __restrict__ W2, int t, int lane,
                                           v2f (&bf)[16]) {
    const int n  = t*16 + (lane & 15);
    const int kb = (lane >> 4) << 1;
    const float* base = W2 + n*MIDC + kb;
#pragma unroll
    for (int j = 0; j < 16; ++j)
        bf[j] = *(const v2f*)(base + j*4);
}

__device__ __forceinline__ v8f zero8() {
    return v8f{0.f,0.f,0.f,0.f,0.f,0.f,0.f,0.f};
}

#define WMMA_F32(af, bf, acc) \
    __builtin_amdgcn_wmma_f32_16x16x4_f32(false, (af), false, (bf), (short)0, (acc), false, false)

// ---------------- K4: conv2 pre-activation stats via fp32 WMMA ----------------
// 8 waves/block: wave = (stripgroup sg in 0..1) x (tilepair tp in 0..3).
// Each wave owns 2 N-tiles with register-resident B-fragments and grid-strides
// over 16-row strips.
__global__ void __launch_bounds__(256) k_conv2_stats(const float* __restrict__ xws,
                                                     const float* __restrict__ W2,
                                                     float* __restrict__ stat) {
    __shared__ float W1Fs[MIDC*3], B1Fs[MIDC];
    __shared__ float ssum[OUTC], ssq[OUTC];
    for (int i = threadIdx.x; i < MIDC*3; i += blockDim.x) W1Fs[i] = stat[iW1F + i];
    for (int i = threadIdx.x; i < MIDC;   i += blockDim.x) B1Fs[i] = stat[iB1F + i];
    for (int i = threadIdx.x; i < OUTC;   i += blockDim.x) { ssum[i] = 0.f; ssq[i] = 0.f; }
    __syncthreads();

    const int lane = threadIdx.x & 31, wv = threadIdx.x >> 5;
    const int tp = wv & 3;          // tile pair: tiles 2*tp, 2*tp+1
    const int sg = wv >> 2;         // strip group 0..1
    const int t0 = 2*tp, t1 = 2*tp + 1;

    v2f bf0[16], bf1[16];
    load_bfrag(W2, t0, lane, bf0);
    load_bfrag(W2, t1, lane, bf1);

    float s10 = 0.f, s20 = 0.f, s11 = 0.f, s21 = 0.f;

#pragma unroll 1
    for (int strip = blockIdx.x*2 + sg; strip < NSTRIP; strip += gridDim.x*2) {
        v2f af[16];
        make_afrag(xws, strip*16, lane, W1Fs, B1Fs, af);
        v8f a0 = zero8(), a1 = zero8();
#pragma unroll
        for (int j = 0; j < 16; ++j) a0 = WMMA_F32(af[j], bf0[j], a0);
#pragma unroll
        for (int j = 0; j < 16; ++j) a1 = WMMA_F32(af[j], bf1[j], a1);
#pragma unroll
        for (int r = 0; r < 8; ++r) {
            s10 += a0[r]; s20 += a0[r]*a0[r];
            s11 += a1[r]; s21 += a1[r]*a1[r];
        }
    }
    // lane l and l+16 hold the same channel n = t*16+(l&15): fold halves, then LDS.
    float u;
    u = s10 + __shfl_xor(s10, 16, 32); if (lane < 16) atomicAdd(&ssum[t0*16 + lane], u);
    u = s20 + __shfl_xor(s20, 16, 32); if (lane < 16) atomicAdd(&ssq [t0*16 + lane], u);
    u = s11 + __shfl_xor(s11, 16, 32); if (lane < 16) atomicAdd(&ssum[t1*16 + lane], u);
    u = s21 + __shfl_xor(s21, 16, 32); if (lane < 16) atomicAdd(&ssq [t1*16 + lane], u);
    __syncthreads();
    for (int i = threadIdx.x; i < OUTC; i += blockDim.x) {
        atomicAdd(&stat[iSUM2 + i], ssum[i]);
        atomicAdd(&stat[iSSQ2 + i], ssq[i]);
    }
}

// ---------------- K5: fold BN2 (b2 cancels exactly) ----------------
__global__ void k_fin2(const float* __restrict__ g2, const float* __restrict__ be2,
                       float* __restrict__ stat) {
    int n = threadIdx.x;
    if (n >= OUTC) return;
    const float Rinv = 1.0f / (float)ROWS;
    float m   = stat[iSUM2 + n] * Rinv;
    float var = stat[iSSQ2 + n] * Rinv - m*m;
    float A   = g2[n] * rsqrtf(var + EPSf);
    stat[iA2  + n] = A;
    stat[iB2F + n] = be2[n] - m*A;
}

// ---------------- K6: conv2 + BN2 + ReLU + max over K, via fp32 WMMA ----------------
// 8 waves/block: wave = (query p in 0..1) x (tilepair tp in 0..3).
// 4 waves cooperate on one (b,s); each owns 2 register-resident B tiles and
// runs both 16-row strips (K=32 neighbors).
__global__ void __launch_bounds__(256) k_conv2_out(const float* __restrict__ xws,
                                                   const float* __restrict__ W2,
                                                   const float* __restrict__ stat,
                                                   float* __restrict__ out) {
    __shared__ float W1Fs[MIDC*3], B1Fs[MIDC], A2s[OUTC], B2s[OUTC];
    for (int i = threadIdx.x; i < MIDC*3; i += blockDim.x) W1Fs[i] = stat[iW1F + i];
    for (int i = threadIdx.x; i < MIDC;   i += blockDim.x) B1Fs[i] = stat[iB1F + i];
    for (int i = threadIdx.x; i < OUTC;   i += blockDim.x) { A2s[i] = stat[iA2 + i]; B2s[i] = stat[iB2F + i]; }
    __syncthreads();

    const int lane = threadIdx.x & 31, wv = threadIdx.x >> 5;
    const int tp = wv & 3;
    const int p  = wv >> 2;                 // query within block
    const int q  = blockIdx.x*2 + p;        // q in [0, B*S)
    const int b  = q >> 11, s = q & (Ss - 1);
    const int nl = lane & 15;
    const int t0 = 2*tp, t1 = 2*tp + 1;

    v2f bf0[16], bf1[16];
    load_bfrag(W2, t0, lane, bf0);
    load_bfrag(W2, t1, lane, bf1);

    const float A0 = A2s[t0*16 + nl], Bc0 = B2s[t0*16 + nl];
    const float A1 = A2s[t1*16 + nl], Bc1 = B2s[t1*16 + nl];

    float tm0 = -3.0e38f, tm1 = -3.0e38f;

#pragma unroll 1
    for (int h = 0; h < 2; ++h) {
        v2f af[16];
        make_afrag(xws, q*Kk + h*16, lane, W1Fs, B1Fs, af);
        v8f a0 = zero8(), a1 = zero8();
#pragma unroll
        for (int j = 0; j < 16; ++j) a0 = WMMA_F32(af[j], bf0[j], a0);
#pragma unroll
        for (int j = 0; j < 16; ++j) a1 = WMMA_F32(af[j], bf1[j], a1);
        float lm0 = -3.0e38f, lm1 = -3.0e38f;
#pragma unroll
        for (int r = 0; r < 8; ++r) {
            lm0 = fmaxf(lm0, fmaxf(a0[r]*A0 + Bc0, 0.0f));
            lm1 = fmaxf(lm1, fmaxf(a1[r]*A1 + Bc1, 0.0f));
        }
        lm0 = fmaxf(lm0, __shfl_xor(lm0, 16, 32));   // combine M halves (same channel)
        lm1 = fmaxf(lm1, __shfl_xor(lm1, 16, 32));
        tm0 = fmaxf(tm0, lm0);
        tm1 = fmaxf(tm1, lm1);
    }
    if (lane < 16) {
        out[FEAT_OFF + ((size_t)(b*OUTC + t0*16 + lane))*Ss + s] = tm0;   // (B, O, S)
        out[FEAT_OFF + ((size_t)(b*OUTC + t1*16 + lane))*Ss + s] = tm1;
    }
}

// ---------------- launch ----------------
extern "C" void kernel_launch(void* const* d_in, const int* in_sizes, int n_in,
                              void* d_out, int out_size, void* d_ws, size_t ws_size,
                              hipStream_t stream) {
    (void)in_sizes; (void)n_in; (void)out_size; (void)ws_size;
    const float* xyz = (const float*)d_in[0];
    const float* W1  = (const float*)d_in[1];
    // d_in[2] = b1 (cancels in folded BN1)
    const float* g1  = (const float*)d_in[3];
    const float* be1 = (const float*)d_in[4];
    const float* W2  = (const float*)d_in[5];
    // d_in[6] = b2 (cancels in folded BN2)
    const float* g2  = (const float*)d_in[7];
    const float* be2 = (const float*)d_in[8];

    float* out  = (float*)d_out;
    float* ws   = (float*)d_ws;
    float* xws  = ws;
    float* stat = ws + XWS_FLOATS;

    k_zero<<<1, 1024, 0, stream>>>(stat);

    size_t sm1 = (size_t)(3*Nn + 96) * sizeof(float);          // ~96.4 KB LDS
    k_fps<<<Bb, 1024, sm1, stream>>>(xyz, out);

    size_t sm2 = (size_t)(3*Nn) * sizeof(float);               // 96 KB LDS
    k_knn_group<<<dim3(Ss/256, Bb), 256, sm2, stream>>>(xyz, out, xws);

    k_moments<<<256, 256, 0, stream>>>(xws, stat);
    k_fin1<<<1, 64, 0, stream>>>(W1, g1, be1, stat);
    k_conv2_stats<<<1024, 256, 0, stream>>>(xws, W2, stat);
    k_fin2<<<1, 128, 0, stream>>>(g2, be2, stat);
    k_conv2_out<<<(Bb*Ss)/2, 256, 0, stream>>>(xws, W2, stat, out);
}